// GaussianPrimitiveBank_3925600109361
// MI455X (gfx1250) — compile-verified
//
#include <hip/hip_runtime.h>
#include <hip/hip_bf16.h>
#include <math.h>
#include <stdint.h>

// Problem constants (match reference): K=512 primitives, D=64, B=4096.
#define NB   4096
#define NK   512
#define ND   64
#define NTRI 2016            // D*(D-1)/2 strict lower
#define NSYM 2080            // D*(D+1)/2 incl. diag
#define KF   2176            // 2080 quad + 64 linear + 1 const + 31 pad
#define BKT  32              // GEMM k-tile
#define BM   128             // block tile M (batch rows)
#define BN   64              // block tile N (primitive cols)
#define LDSS 36              // LDS row stride: 16B-aligned b128 rows, conflict-free b64 reads

#define USE_ASYNC_COPY 1

typedef __attribute__((ext_vector_type(2))) float v2f;
typedef __attribute__((ext_vector_type(8))) float v8f;
typedef __attribute__((address_space(3))) float lds_f;

#ifdef __has_builtin
#if __has_builtin(__builtin_amdgcn_s_wait_asynccnt)
#define GPB_WAIT_ASYNC0() __builtin_amdgcn_s_wait_asynccnt(0)
#endif
#endif
#ifndef GPB_WAIT_ASYNC0
#define GPB_WAIT_ASYNC0() asm volatile("s_wait_asynccnt 0x0" ::: "memory")
#endif

#if USE_ASYNC_COPY
// ASYNCcnt-tracked 16B/lane copy: copies from the global address g to the LDS address lds_off, no VGPR data staging.
__device__ __forceinline__ void gpb_async_b128(uint32_t lds_off, const float* g) {
  asm volatile("global_load_async_to_lds_b128 %0, %1, off"
               :: "v"(lds_off), "v"((unsigned long long)(uintptr_t)g)
               : "memory");
}
#endif

// ---------------------------------------------------------------------------
// Kernel 1: per-batch feature rows Q[b][KF]
//   f < 2080 : f = i*(i+1)/2 + j (j<=i):  i==j ? z_i^2 : 2*z_i*z_j
//   2080+d   : z_d ; 2144 : 1.0 ; pad : 0
// ---------------------------------------------------------------------------
__global__ __launch_bounds__(256) void gpb_prep_q(const float* __restrict__ z,
                                                  float* __restrict__ Q) {
  const int b = blockIdx.x;
  const int t = threadIdx.x;
  __shared__ float zs[ND];
  if (t < ND) zs[t] = z[b * ND + t];
  __syncthreads();
  const size_t base = (size_t)b * KF;
  for (int f = t; f < KF; f += 256) {
    float val;
    if (f < NSYM) {
      int i = (int)((sqrtf(8.0f * (float)f + 1.0f) - 1.0f) * 0.5f);
      while ((i + 1) * (i + 2) / 2 <= f) ++i;
      while (i * (i + 1) / 2 > f) --i;
      int j = f - i * (i + 1) / 2;
      float zi = zs[i], zj = zs[j];
      val = (i == j) ? zi * zi : 2.0f * zi * zj;
    } else if (f < NSYM + ND) {
      val = zs[f - NSYM];
    } else if (f == NSYM + ND) {
      val = 1.0f;
    } else {
      val = 0.0f;
    }
    Q[base + f] = val;
  }
}

// ---------------------------------------------------------------------------
// Kernel 2: per-primitive column S[k][KF] (K-major == B^T for the GEMM)
//   A = L L^T, v = A*mu, c = mu^T v, logw = log(softplus(log_s)+eps)
//   f<2080: -0.5*A[i][j] ; 2080+d: v_d ; 2144: -0.5c+logw ; pad: 0
// ---------------------------------------------------------------------------
__global__ __launch_bounds__(256) void gpb_prep_s(const float* __restrict__ mu,
                                                  const float* __restrict__ log_diag,
                                                  const float* __restrict__ log_s,
                                                  const float* __restrict__ lod,
                                                  float* __restrict__ S) {
  const int k = blockIdx.x;
  const int t = threadIdx.x;
  __shared__ float L[ND][ND + 1];
  __shared__ float Af[ND][ND + 1];
  __shared__ float mus[ND];
  __shared__ float vv[ND];
  float* Lf = &L[0][0];
  for (int idx = t; idx < ND * (ND + 1); idx += 256) Lf[idx] = 0.0f;
  __syncthreads();
  if (t < ND) {
    L[t][t] = expf(0.5f * log_diag[k * ND + t]);
    mus[t]  = mu[k * ND + t];
  }
  for (int idx = t; idx < NTRI; idx += 256) {       // tril_indices(D,-1) row-major
    int i = (int)((1.0f + sqrtf(1.0f + 8.0f * (float)idx)) * 0.5f);
    while (i * (i - 1) / 2 > idx) --i;
    while ((i + 1) * i / 2 <= idx) ++i;
    int j = idx - i * (i - 1) / 2;
    L[i][j] = lod[k * NTRI + idx];
  }
  __syncthreads();
  const size_t base = (size_t)k * KF;
  for (int f = t; f < NSYM; f += 256) {
    int i = (int)((sqrtf(8.0f * (float)f + 1.0f) - 1.0f) * 0.5f);
    while ((i + 1) * (i + 2) / 2 <= f) ++i;
    while (i * (i + 1) / 2 > f) --i;
    int j = f - i * (i + 1) / 2;                    // j <= i
    float acc = 0.0f;
    for (int d = 0; d <= j; ++d) acc += L[i][d] * L[j][d];
    Af[i][j] = acc;
    Af[j][i] = acc;
    S[base + f] = -0.5f * acc;
  }
  __syncthreads();
  if (t < ND) {
    float acc = 0.0f;
    for (int j = 0; j < ND; ++j) acc += Af[t][j] * mus[j];
    vv[t] = acc;
    S[base + NSYM + t] = acc;
  }
  __syncthreads();
  if (t == 0) {
    float c = 0.0f;
    for (int j = 0; j < ND; ++j) c += mus[j] * vv[j];
    float ls = log_s[k];
    float sp = (ls > 20.0f) ? ls : log1pf(expf(ls));
    S[base + NSYM + ND] = -0.5f * c + logf(sp + 1e-8f);
  }
  for (int f = NSYM + ND + 1 + t; f < KF; f += 256) S[base + f] = 0.0f;
}

// ---------------------------------------------------------------------------
// Kernel 3: log_a (4096x512) = Q (4096xKF) * S^T via V_WMMA_F32_16X16X4_F32
// 256 thr = 8 waves (4M x 2N), wave tile 32x32 (4 accums), block tile 128x64,
// double-buffered LDS filled by GLOBAL_LOAD_ASYNC_TO_LDS_B128 (ASYNCcnt).
// ---------------------------------------------------------------------------
__global__ __launch_bounds__(256) void gpb_gemm_loga(const float* __restrict__ Q,
                                                     const float* __restrict__ S,
                                                     float* __restrict__ out) {
  __shared__ float Qs[2][BM][LDSS];
  __shared__ float Ss[2][BN][LDSS];

  const int tid  = threadIdx.x;
  const int lane = tid & 31;
  const int wid  = tid >> 5;
  const int wm   = wid & 3;              // 0..3 -> 32 M rows each
  const int wn   = wid >> 2;             // 0..1 -> 32 N cols each
  const int rowBase = blockIdx.x * BM;   // over B
  const int colBase = blockIdx.y * BN;   // over K(outputs)

  // Copy mapping: 8 b128 per 32-float row slice; thread -> (row rB + 32*i, col c4).
  const int rB = tid >> 3;
  const int c4 = (tid & 7) * 4;
  const float* qbase = Q + (size_t)(rowBase + rB) * KF + c4;
  const float* sbase = S + (size_t)(colBase + rB) * KF + c4;

#if USE_ASYNC_COPY
  const uint32_t qb   = (uint32_t)(size_t)(lds_f*)&Qs[0][rB][c4];
  const uint32_t sb   = (uint32_t)(size_t)(lds_f*)&Ss[0][rB][c4];
  const uint32_t QBUF = BM * LDSS * 4;
  const uint32_t SBUF = BN * LDSS * 4;
  const uint32_t R32  = 32 * LDSS * 4;

  // prefetch chunk 0 into buffer 0
  gpb_async_b128(qb,           qbase);
  gpb_async_b128(qb + R32,     qbase + 32 * KF);
  gpb_async_b128(qb + 2 * R32, qbase + 64 * KF);
  gpb_async_b128(qb + 3 * R32, qbase + 96 * KF);
  gpb_async_b128(sb,           sbase);
  gpb_async_b128(sb + R32,     sbase + 32 * KF);
  GPB_WAIT_ASYNC0();
#else
  float4 rq[4], rs[2];
#pragma unroll
  for (int i = 0; i < 4; ++i) rq[i] = *(const float4*)(qbase + 32 * i * KF);
#pragma unroll
  for (int i = 0; i < 2; ++i) rs[i] = *(const float4*)(sbase + 32 * i * KF);
#pragma unroll
  for (int i = 0; i < 4; ++i) {
    Qs[0][rB + 32 * i][c4 + 0] = rq[i].x; Qs[0][rB + 32 * i][c4 + 1] = rq[i].y;
    Qs[0][rB + 32 * i][c4 + 2] = rq[i].z; Qs[0][rB + 32 * i][c4 + 3] = rq[i].w;
  }
#pragma unroll
  for (int i = 0; i < 2; ++i) {
    Ss[0][rB + 32 * i][c4 + 0] = rs[i].x; Ss[0][rB + 32 * i][c4 + 1] = rs[i].y;
    Ss[0][rB + 32 * i][c4 + 2] = rs[i].z; Ss[0][rB + 32 * i][c4 + 3] = rs[i].w;
  }
#endif
  __syncthreads();

  v8f c00 = {}, c01 = {}, c10 = {}, c11 = {};

  // Fragment addressing (ISA 7.12.2): lane l, vgpr j -> (row=l&15, K=2*(l>>4)+j)
  const int aRow = wm * 32 + (lane & 15);
  const int bRow = wn * 32 + (lane & 15);
  const int kOff = 2 * (lane >> 4);

  const int NCH = KF / BKT;   // 68
  for (int ch = 0; ch < NCH; ++ch) {
    const int cur   = ch & 1;
    const bool more = (ch + 1) < NCH;
    if (more) {  // async copy of next chunk overlaps the WMMA phase below
      const int k0 = (ch + 1) * BKT;
#if USE_ASYNC_COPY
      const uint32_t nxt = (uint32_t)(cur ^ 1);
      const uint32_t qo = qb + nxt * QBUF;
      const uint32_t so = sb + nxt * SBUF;
      gpb_async_b128(qo,           qbase + k0);
      gpb_async_b128(qo + R32,     qbase + 32 * KF + k0);
      gpb_async_b128(qo + 2 * R32, qbase + 64 * KF + k0);
      gpb_async_b128(qo + 3 * R32, qbase + 96 * KF + k0);
      gpb_async_b128(so,           sbase + k0);
      gpb_async_b128(so + R32,     sbase + 32 * KF + k0);
#else
#pragma unroll
      for (int i = 0; i < 4; ++i) rq[i] = *(const float4*)(qbase + 32 * i * KF + k0);
#pragma unroll
      for (int i = 0; i < 2; ++i) rs[i] = *(const float4*)(sbase + 32 * i * KF + k0);
#endif
    }
#pragma unroll
    for (int kk = 0; kk < BKT; kk += 4) {
      v2f a0 = *(const v2f*)&Qs[cur][aRow][kk + kOff];
      v2f a1 = *(const v2f*)&Qs[cur][aRow + 16][kk + kOff];
      v2f b0 = *(const v2f*)&Ss[cur][bRow][kk + kOff];
      v2f b1 = *(const v2f*)&Ss[cur][bRow + 16][kk + kOff];
      c00 = __builtin_amdgcn_wmma_f32_16x16x4_f32(false, a0, false, b0, (short)0, c00, false, false);
      c01 = __builtin_amdgcn_wmma_f32_16x16x4_f32(false, a0, false, b1, (short)0, c01, false, false);
      c10 = __builtin_amdgcn_wmma_f32_16x16x4_f32(false, a1, false, b0, (short)0, c10, false, false);
      c11 = __builtin_amdgcn_wmma_f32_16x16x4_f32(false, a1, false, b1, (short)0, c11, false, false);
    }
    if (more) {
#if USE_ASYNC_COPY
      GPB_WAIT_ASYNC0();          // own async writes done before the barrier
#else
      const int nxt = cur ^ 1;
#pragma unroll
      for (int i = 0; i < 4; ++i) {
        Qs[nxt][rB + 32 * i][c4 + 0] = rq[i].x; Qs[nxt][rB + 32 * i][c4 + 1] = rq[i].y;
        Qs[nxt][rB + 32 * i][c4 + 2] = rq[i].z; Qs[nxt][rB + 32 * i][c4 + 3] = rq[i].w;
      }
#pragma unroll
      for (int i = 0; i < 2; ++i) {
        Ss[nxt][rB + 32 * i][c4 + 0] = rs[i].x; Ss[nxt][rB + 32 * i][c4 + 1] = rs[i].y;
        Ss[nxt][rB + 32 * i][c4 + 2] = rs[i].z; Ss[nxt][rB + 32 * i][c4 + 3] = rs[i].w;
      }
#endif
    }
    __syncthreads();
  }

  // C layout: vgpr p -> (M = p + 8*(l>>4), N = l&15)
  const int om = rowBase + wm * 32 + 8 * (lane >> 4);
  const int on = colBase + wn * 32 + (lane & 15);
#pragma unroll
  for (int p = 0; p < 8; ++p) {
    out[(size_t)(om + p) * NK + on]           = c00[p];
    out[(size_t)(om + p) * NK + on + 16]      = c01[p];
    out[(size_t)(om + 16 + p) * NK + on]      = c10[p];
    out[(size_t)(om + 16 + p) * NK + on + 16] = c11[p];
  }
}

// ---------------------------------------------------------------------------
// Kernel 4: in-place row softmax over K=512 (one wave32 per row, 16 vals/lane)
// ---------------------------------------------------------------------------
__global__ __launch_bounds__(256) void gpb_softmax(float* __restrict__ out) {
  const int lane = threadIdx.x & 31;
  const int wid  = threadIdx.x >> 5;
  const int row  = blockIdx.x * 8 + wid;
  float* p = out + (size_t)row * NK;
  float v[16];
#pragma unroll
  for (int i = 0; i < 16; ++i) v[i] = p[lane + i * 32];
  float m = v[0];
#pragma unroll
  for (int i = 1; i < 16; ++i) m = fmaxf(m, v[i]);
#pragma unroll
  for (int off = 16; off > 0; off >>= 1) m = fmaxf(m, __shfl_xor(m, off, 32));
  float s = 0.0f;
#pragma unroll
  for (int i = 0; i < 16; ++i) { v[i] = expf(v[i] - m); s += v[i]; }
#pragma unroll
  for (int off = 16; off > 0; off >>= 1) s += __shfl_xor(s, off, 32);
  const float inv = 1.0f / (s + 1e-8f);
#pragma unroll
  for (int i = 0; i < 16; ++i) p[lane + i * 32] = v[i] * inv;
}

// ---------------------------------------------------------------------------
extern "C" void kernel_launch(void* const* d_in, const int* in_sizes, int n_in,
                              void* d_out, int out_size, void* d_ws, size_t ws_size,
                              hipStream_t stream) {
  (void)in_sizes; (void)n_in; (void)out_size; (void)ws_size;
  const float* z        = (const float*)d_in[0];   // (B, D)
  const float* mu       = (const float*)d_in[1];   // (K, D)
  const float* log_diag = (const float*)d_in[2];   // (K, D)
  const float* log_s    = (const float*)d_in[3];   // (K,)
  const float* lod      = (const float*)d_in[4];   // (K, D*(D-1)/2)
  float* out = (float*)d_out;                      // (B, K)

  float* Q = (float*)d_ws;                         // B * KF floats (~34 MB)
  float* S = Q + (size_t)NB * KF;                  // K * KF floats (~4.3 MB)

  gpb_prep_q<<<NB, 256, 0, stream>>>(z, Q);
  gpb_prep_s<<<NK, 256, 0, stream>>>(mu, log_diag, log_s, lod, S);
  dim3 grid(NB / BM, NK / BN);                     // 32 x 8 blocks
  gpb_gemm_loga<<<grid, 256, 0, stream>>>(Q, S, out);
  gpb_softmax<<<NB / 8, 256, 0, stream>>>(out);
}